// Net_62182536511754
// MI455X (gfx1250) — compile-verified
//
#include <hip/hip_runtime.h>
#include <math.h>

#define N_NODES 50000
#define N_EDGES 800000
#define K_IN 128

typedef __attribute__((ext_vector_type(2))) float v2f;
typedef __attribute__((ext_vector_type(8))) float v8f;

// Monotonic float <-> uint mapping so atomicMax(u32) implements float segment-max.
static __device__ __forceinline__ unsigned ford(float v) {
  unsigned b = __float_as_uint(v);
  return (b & 0x80000000u) ? ~b : (b | 0x80000000u);
}
static __device__ __forceinline__ float ford_inv(unsigned b) {
  b = (b & 0x80000000u) ? (b & 0x7FFFFFFFu) : ~b;
  return __uint_as_float(b);
}
#define ORD_NEG_INF 0x007FFFFFu  // ford(-inf)

// ---------------------------------------------------------------------------
// GEMM: Y[N,LDY] = X[N,128] @ W[128,FOUT], fp32 via V_WMMA_F32_16X16X4_F32.
// FOUT = NT*16 compile-time; W pre-interleaved in K-pairs:
//   WP[((k>>1)*FOUT + c)*2 + (k&1)] = W[k][c]
// so each B fragment is ONE aligned b64 load at a constant immediate offset.
// A frag (16x4 f32): lane&15 = row, lane>>4 selects K pair {0,1} vs {2,3}.
// C/D (16x16 f32):  vgpr r -> row r (lanes 0-15) / row 8+r (lanes 16-31).
// ---------------------------------------------------------------------------
template <int NT, int LDY>
__global__ __launch_bounds__(256) void gemm_wmma(const float* __restrict__ X,
                                                 const float* __restrict__ WP,
                                                 float* __restrict__ Y) {
  constexpr int FOUT = NT * 16;
  const int wave = (blockIdx.x * blockDim.x + threadIdx.x) >> 5;
  const int lane = threadIdx.x & 31;
  if (wave >= (N_NODES / 16)) return;  // wave-uniform exit
  const int m0 = wave * 16;
  const int mr = lane & 15;
  const int khalf = lane >> 4;  // 0/1 -> K offset 0/2 within each group of 4

  v8f acc[NT];
#pragma unroll
  for (int t = 0; t < NT; ++t) acc[t] = (v8f)0.0f;

  const float* abase = X + (size_t)(m0 + mr) * K_IN + khalf * 2;
  const float* bbase = WP + ((size_t)khalf * FOUT + mr) * 2;

#pragma unroll
  for (int k0 = 0; k0 < K_IN; k0 += 4) {
    v2f a = *(const v2f*)(abase + k0);  // K = k0+2*khalf, +1
#pragma unroll
    for (int t = 0; t < NT; ++t) {
      v2f b = *(const v2f*)(bbase + (size_t)k0 * FOUT + t * 32);
      acc[t] = __builtin_amdgcn_wmma_f32_16x16x4_f32(
          false, a, false, b, (short)0, acc[t], false, false);
    }
  }

  const int rbase = khalf * 8;
  float* ybase = Y + (size_t)(m0 + rbase) * LDY + mr;
#pragma unroll
  for (int t = 0; t < NT; ++t) {
    const int c = t * 16 + mr;
    if (c < LDY) {  // only partial for the padded last tile (LDY=40)
#pragma unroll
      for (int r = 0; r < 8; ++r)
        ybase[(size_t)r * LDY + t * 16] = acc[t][r];
    }
  }
}

// Interleave W[128][Fsrc] (zero-padded to Fdst cols) into K-paired layout.
__global__ void pack_pairs(const float* __restrict__ W, float* __restrict__ WP,
                           int Fsrc, int Fdst) {
  int i = blockIdx.x * blockDim.x + threadIdx.x;
  if (i >= 128 * Fdst) return;
  int k = i / Fdst, c = i - k * Fdst;
  float v = (c < Fsrc) ? W[(size_t)k * Fsrc + c] : 0.f;
  WP[(((size_t)(k >> 1)) * Fdst + c) * 2 + (k & 1)] = v;
}

// W_heads[4][128][32] -> K-paired concat layout (columns = head-major concat).
__global__ void pack_heads_pairs(const float* __restrict__ Wh, float* __restrict__ WP) {
  int i = blockIdx.x * blockDim.x + threadIdx.x;
  if (i >= 128 * 128) return;
  int k = i >> 7, c = i & 127;
  int h = c >> 5, j = c & 31;
  float v = Wh[((size_t)h * 128 + k) * 32 + j];
  WP[(((size_t)(k >> 1)) * 128 + c) * 2 + (k & 1)] = v;
}

// Per-node attention score halves: EC[h*N+n] = Wh_row . a_center, EN = . a_neigh
__global__ void scores_kernel(const float* __restrict__ Wh, const float* __restrict__ a,
                              float* __restrict__ EC, float* __restrict__ EN,
                              int H, int Fh, int Frow) {
  int idx = blockIdx.x * blockDim.x + threadIdx.x;
  if (idx >= N_NODES * H) return;
  int h = idx / N_NODES;
  int n = idx - h * N_NODES;
  const float* row = Wh + (size_t)n * Frow + h * Fh;
  const float* ac = a + (size_t)h * 2 * Fh;
  const float* an = ac + Fh;
  float sc = 0.f, sn = 0.f;
  for (int j = 0; j < Fh; ++j) {
    float v = row[j];
    sc = fmaf(v, ac[j], sc);
    sn = fmaf(v, an[j], sn);
  }
  EC[idx] = sc;
  EN[idx] = sn;
}

__global__ void fill_u32(unsigned* __restrict__ p, unsigned v, int n) {
  int i = blockIdx.x * blockDim.x + threadIdx.x;
  if (i < n) p[i] = v;
}

// Pass 1: e = leakyrelu(ec[dst]+en[src]); segment max via ordered-uint atomicMax.
__global__ void edge_max_kernel(const int* __restrict__ src, const int* __restrict__ dst,
                                const float* __restrict__ EC, const float* __restrict__ EN,
                                float* __restrict__ EDG, unsigned* __restrict__ MB, int H) {
  int idx = blockIdx.x * blockDim.x + threadIdx.x;
  if (idx >= N_EDGES * H) return;
  int h = idx / N_EDGES;
  int e = idx - h * N_EDGES;
  int s = src[e], d = dst[e];
  float v = EC[(size_t)h * N_NODES + d] + EN[(size_t)h * N_NODES + s];
  v = (v > 0.f) ? v : 0.2f * v;  // LeakyReLU(0.2)
  EDG[idx] = v;
  atomicMax(&MB[(size_t)h * N_NODES + d], ford(v));
}

// Pass 2: ex = exp(e - max[dst]); segment sum into DEN.
__global__ void edge_exp_kernel(const int* __restrict__ dst, float* __restrict__ EDG,
                                const unsigned* __restrict__ MB, float* __restrict__ DEN,
                                int H) {
  int idx = blockIdx.x * blockDim.x + threadIdx.x;
  if (idx >= N_EDGES * H) return;
  int h = idx / N_EDGES;
  int e = idx - h * N_EDGES;
  int d = dst[e];
  float m = ford_inv(MB[(size_t)h * N_NODES + d]);
  float ex = expf(EDG[idx] - m);
  EDG[idx] = ex;
  atomicAdd(&DEN[(size_t)h * N_NODES + d], ex);
}

// Pass 3: AGG[dst] += (ex/denom[dst]) * Wh[src].  One wave per (head, edge).
__global__ __launch_bounds__(256) void edge_agg_kernel(
    const int* __restrict__ src, const int* __restrict__ dst,
    const float* __restrict__ EDG, const float* __restrict__ DEN,
    const float* __restrict__ Wh, float* __restrict__ AGG, int H, int Fh, int Frow) {
  int gid = blockIdx.x * 8 + (threadIdx.x >> 5);
  int lane = threadIdx.x & 31;
  if (gid >= N_EDGES * H) return;  // wave-uniform
  int h = gid / N_EDGES;
  int e = gid - h * N_EDGES;
  int s = src[e], d = dst[e];
  float coef = EDG[gid] / fmaxf(DEN[(size_t)h * N_NODES + d], 1e-9f);
  const float* wrow = Wh + (size_t)s * Frow + h * Fh;
  float* orow = AGG + (size_t)d * Frow + h * Fh;
  for (int f = lane; f < Fh; f += 32)
    atomicAdd(&orow[f], coef * wrow[f]);
}

// mode 0: ELU(alpha=1), mode 1: ReLU
__global__ void act_kernel(const float* __restrict__ in, float* __restrict__ out,
                           int n, int mode) {
  int i = blockIdx.x * blockDim.x + threadIdx.x;
  if (i >= n) return;
  float v = in[i];
  out[i] = (mode == 0) ? ((v > 0.f) ? v : (expf(v) - 1.f)) : fmaxf(v, 0.f);
}

__global__ void logsoftmax_kernel(const float* __restrict__ in, float* __restrict__ out) {
  int n = blockIdx.x * blockDim.x + threadIdx.x;
  if (n >= N_NODES) return;
  const float* row = in + (size_t)n * 40;
  float m = -1e30f;
  for (int j = 0; j < 40; ++j) m = fmaxf(m, row[j]);
  float s = 0.f;
  for (int j = 0; j < 40; ++j) s += expf(row[j] - m);
  float ls = m + logf(s);
  float* orow = out + (size_t)n * 40;
  for (int j = 0; j < 40; ++j) orow[j] = row[j] - ls;
}

// ---------------------------------------------------------------------------

static void gat_layer_post(hipStream_t stream, const float* a, int Fout, int H, int Fh,
                           const int* src, const int* dst, float* Wh, float* AGG,
                           float* EC, float* EN, unsigned* MB, float* DEN, float* EDG) {
  const int Frow = Fout;  // = H * Fh
  {
    int n = N_NODES * H;
    scores_kernel<<<(n + 255) / 256, 256, 0, stream>>>(Wh, a, EC, EN, H, Fh, Frow);
    fill_u32<<<(n + 255) / 256, 256, 0, stream>>>(MB, ORD_NEG_INF, n);
    fill_u32<<<(n + 255) / 256, 256, 0, stream>>>((unsigned*)DEN, 0u, n);
  }
  {
    int n = N_NODES * Frow;
    fill_u32<<<(n + 255) / 256, 256, 0, stream>>>((unsigned*)AGG, 0u, n);
  }
  int eh = N_EDGES * H;
  edge_max_kernel<<<(eh + 255) / 256, 256, 0, stream>>>(src, dst, EC, EN, EDG, MB, H);
  edge_exp_kernel<<<(eh + 255) / 256, 256, 0, stream>>>(dst, EDG, MB, DEN, H);
  edge_agg_kernel<<<(eh + 7) / 8, 256, 0, stream>>>(src, dst, EDG, DEN, Wh, AGG, H, Fh, Frow);
}

extern "C" void kernel_launch(void* const* d_in, const int* in_sizes, int n_in,
                              void* d_out, int out_size, void* d_ws, size_t ws_size,
                              hipStream_t stream) {
  (void)in_sizes; (void)n_in; (void)out_size; (void)ws_size;
  const float* x       = (const float*)d_in[0];  // [N,128]
  const int*   edge    = (const int*)d_in[1];    // [2,E] (int per harness convention)
  const float* W_heads = (const float*)d_in[2];  // [4,128,32]
  const float* a_heads = (const float*)d_in[3];  // [4,64]
  const float* W_mid   = (const float*)d_in[4];  // [128,128]
  const float* a_mid   = (const float*)d_in[5];  // [256]
  const float* W_out   = (const float*)d_in[6];  // [128,40]
  const float* a_out   = (const float*)d_in[7];  // [80]
  const int* src = edge;
  const int* dst = edge + N_EDGES;

  float* base = (float*)d_ws;
  float* B0    = base;                             // N*128  Wh of current layer
  float* B1    = B0 + (size_t)N_NODES * 128;       // N*128  aggregation buffer
  float* B2    = B1 + (size_t)N_NODES * 128;       // N*128  activated layer input
  float* EC    = B2 + (size_t)N_NODES * 128;       // N*4
  float* EN    = EC + (size_t)N_NODES * 4;         // N*4
  unsigned* MB = (unsigned*)(EN + (size_t)N_NODES * 4);  // N*4
  float* DEN   = (float*)(MB + (size_t)N_NODES * 4);     // N*4
  float* EDG   = DEN + (size_t)N_NODES * 4;        // E*4
  float* WP1   = EDG + (size_t)N_EDGES * 4;        // 128*128 paired layer-1 weights
  float* WP2   = WP1 + 128 * 128;                  // 128*128 paired mid weights
  float* WP3   = WP2 + 128 * 128;                  // 128*48  paired out weights (padded)

  const int gemm_blocks = ((N_NODES / 16) + 7) / 8;  // 8 waves/block

  // Weight packing (paired-K, head-concat / zero-pad as needed).
  pack_heads_pairs<<<64, 256, 0, stream>>>(W_heads, WP1);
  pack_pairs<<<64, 256, 0, stream>>>(W_mid, WP2, 128, 128);
  pack_pairs<<<(128 * 48 + 255) / 256, 256, 0, stream>>>(W_out, WP3, 40, 48);

  // Layer 1: 4 heads fused as one 128x128 GEMM; concat + ELU.
  gemm_wmma<8, 128><<<gemm_blocks, 256, 0, stream>>>(x, WP1, B0);
  gat_layer_post(stream, a_heads, 128, 4, 32, src, dst, B0, B1, EC, EN, MB, DEN, EDG);
  act_kernel<<<(N_NODES * 128 + 255) / 256, 256, 0, stream>>>(B1, B2, N_NODES * 128, 0);

  // Layer 2: single head 128->128, ReLU.
  gemm_wmma<8, 128><<<gemm_blocks, 256, 0, stream>>>(B2, WP2, B0);
  gat_layer_post(stream, a_mid, 128, 1, 128, src, dst, B0, B1, EC, EN, MB, DEN, EDG);
  act_kernel<<<(N_NODES * 128 + 255) / 256, 256, 0, stream>>>(B1, B2, N_NODES * 128, 1);

  // Layer 3: single head 128->40 (padded to 48-col GEMM), then log-softmax.
  gemm_wmma<3, 40><<<gemm_blocks, 256, 0, stream>>>(B2, WP3, B0);
  gat_layer_post(stream, a_out, 40, 1, 40, src, dst, B0, B1, EC, EN, MB, DEN, EDG);
  logsoftmax_kernel<<<(N_NODES + 255) / 256, 256, 0, stream>>>(B1, (float*)d_out);
}